// LatentGeneExpressionGNN_63660005261872
// MI455X (gfx1250) — compile-verified
//
#include <hip/hip_runtime.h>

#define N_NODES 50000
#define FDIM    128
#define HDIM    128
#define N_EDGES 800000
#define G_GRAPHS 512
#define CE_DIM  64
#define LDIM    128
#define LN_EPS  1e-5f

typedef __attribute__((ext_vector_type(2))) float v2f;
typedef __attribute__((ext_vector_type(8))) float v8f;

__device__ __forceinline__ void atomic_add_f32(float* p, float v) {
  __hip_atomic_fetch_add(p, v, __ATOMIC_RELAXED, __HIP_MEMORY_SCOPE_AGENT);
}

__global__ void set_f32(float* __restrict__ p, float val, int n) {
  int i = blockIdx.x * blockDim.x + threadIdx.x;
  if (i < n) p[i] = val;
}

__global__ void edge_degree(const int* __restrict__ dst, float* __restrict__ deg, int ne) {
  int e = blockIdx.x * blockDim.x + threadIdx.x;
  if (e < ne) atomic_add_f32(deg + dst[e], 1.0f);
}

__global__ void rsqrt_inplace(float* __restrict__ p, int n) {
  int i = blockIdx.x * blockDim.x + threadIdx.x;
  if (i < n) p[i] = rsqrtf(p[i]);
}

// C[ntiles*16 x 128] = A[ntiles*16 x 128] * W[128 x 128], fp32 via v_wmma_f32_16x16x4_f32.
// 256 threads = 8 waves; each wave computes one 16x128 output strip.
// W is pre-swizzled into LDS in B-fragment order so each fragment is one
// conflict-free ds_load_b64 landing directly in an even-aligned VGPR pair.
__global__ __launch_bounds__(256) void gemm128_wmma(
    const float* __restrict__ A, const float* __restrict__ W,
    float* __restrict__ C, int ntiles) {
  // B fragment for (kstep, n) in lane L:
  //   half = L>>4, l = L&15, row = 4*kstep + 2*half, col = n*16 + l
  //   pair = ( W[row][col], W[row+1][col] ) stored at v2f index (kstep*8+n)*32 + L
  __shared__ v2f lw[32 * 8 * 32];   // 64 KB
  for (int idx = threadIdx.x; idx < 32 * 8 * 32; idx += 256) {
    int ln    = idx & 31;
    int n     = (idx >> 5) & 7;
    int kstep = idx >> 8;
    int row   = 4 * kstep + 2 * (ln >> 4);
    int col   = n * 16 + (ln & 15);
    v2f p;
    p.x = W[row * HDIM + col];
    p.y = W[(row + 1) * HDIM + col];
    lw[idx] = p;
  }
  __syncthreads();

  int wave = threadIdx.x >> 5;
  int lane = threadIdx.x & 31;
  int tile = blockIdx.x * 8 + wave;
  if (tile >= ntiles) return;

  int halfw = lane >> 4;          // 0: K slots 0,1 ; 1: K slots 2,3
  int l = lane & 15;              // M (for A rows)
  const float* arow = A + (size_t)(tile * 16 + l) * HDIM + 2 * halfw;
  const v2f* bfrag = lw + lane;

  v8f acc[8] = {};
#pragma unroll 4
  for (int kstep = 0; kstep < 32; ++kstep) {
    v2f a = *(const v2f*)(arow + 4 * kstep);   // A[M=l][K=4k+2h], A[M=l][K=4k+2h+1]
#pragma unroll
    for (int n = 0; n < 8; ++n) {
      v2f b = bfrag[(kstep * 8 + n) * 32];     // ds_load_b64, conflict-free
      acc[n] = __builtin_amdgcn_wmma_f32_16x16x4_f32(
          false, a, false, b, (short)0, acc[n], false, false);
    }
  }

  // C/D layout: VGPR r, lanes 0-15 -> M=r, lanes 16-31 -> M=r+8, N = n*16 + l
  int rbase = tile * 16 + 8 * halfw;
#pragma unroll
  for (int n = 0; n < 8; ++n) {
#pragma unroll
    for (int r = 0; r < 8; ++r) {
      C[(size_t)(rbase + r) * HDIM + n * 16 + l] = acc[n][r];
    }
  }
}

// Initialize aggregation buffer with the self-loop term: out[i] = h[i] * dinv[node]^2
__global__ void selfloop_init(const float* __restrict__ h, const float* __restrict__ dinv,
                              float* __restrict__ out, int n) {
  int i = blockIdx.x * blockDim.x + threadIdx.x;
  if (i >= n) return;
  float di = dinv[i >> 7];
  out[i] = h[i] * di * di;
}

// One wave per edge: gather h[src] (float4/lane), scale by dinv[s]*dinv[d],
// hardware-atomic scatter-add into out[dst].
__global__ __launch_bounds__(256) void edge_aggregate(
    const int* __restrict__ src, const int* __restrict__ dst,
    const float* __restrict__ dinv, const float* __restrict__ h,
    float* __restrict__ out, int ne) {
  int e = blockIdx.x * 8 + (threadIdx.x >> 5);
  int lane = threadIdx.x & 31;
  if (e >= ne) return;
  int s = src[e], d = dst[e];
  float nrm = dinv[s] * dinv[d];
  float4 v = ((const float4*)(h + (size_t)s * HDIM))[lane];
  float* o = out + (size_t)d * HDIM + lane * 4;
  atomic_add_f32(o + 0, v.x * nrm);
  atomic_add_f32(o + 1, v.y * nrm);
  atomic_add_f32(o + 2, v.z * nrm);
  atomic_add_f32(o + 3, v.w * nrm);
}

// agg = relu(agg + bias)
__global__ void bias_relu(const float* __restrict__ bias, float* __restrict__ agg, int n) {
  int i = blockIdx.x * blockDim.x + threadIdx.x;
  if (i >= n) return;
  agg[i] = fmaxf(agg[i] + bias[i & 127], 0.0f);
}

// One wave per node: atomic add row into gpool[batch[node]]
__global__ __launch_bounds__(256) void pool_add(
    const float* __restrict__ h, const int* __restrict__ batch,
    float* __restrict__ g, int nn) {
  int node = blockIdx.x * 8 + (threadIdx.x >> 5);
  int lane = threadIdx.x & 31;
  if (node >= nn) return;
  int b = batch[node];
  float4 v = ((const float4*)(h + (size_t)node * HDIM))[lane];
  float* o = g + (size_t)b * HDIM + lane * 4;
  atomic_add_f32(o + 0, v.x);
  atomic_add_f32(o + 1, v.y);
  atomic_add_f32(o + 2, v.z);
  atomic_add_f32(o + 3, v.w);
}

__device__ __forceinline__ float blocksum128(float v, float* red, int t) {
  __syncthreads();          // protect previous use of red
  red[t] = v;
  __syncthreads();
#pragma unroll
  for (int s = 64; s > 0; s >>= 1) {
    if (t < s) red[t] += red[t + s];
    __syncthreads();
  }
  return red[0];
}

// Fused head: concat(g, emb[cell]) @ Wc + bc -> LN -> relu -> @Wf1 + bf1 -> relu -> LN -> @Wf2 + bf2
__global__ __launch_bounds__(128) void head_kernel(
    const float* __restrict__ g, const int* __restrict__ cell,
    const float* __restrict__ emb,
    const float* __restrict__ Wc, const float* __restrict__ bc,
    const float* __restrict__ lncw, const float* __restrict__ lncb,
    const float* __restrict__ Wf1, const float* __restrict__ bf1,
    const float* __restrict__ ln1w, const float* __restrict__ ln1b,
    const float* __restrict__ Wf2, const float* __restrict__ bf2,
    float* __restrict__ out) {
  int gi = blockIdx.x, t = threadIdx.x;
  __shared__ float comb[HDIM + CE_DIM];
  __shared__ float red[128];
  __shared__ float vec[128];

  comb[t] = g[gi * HDIM + t];
  if (t < CE_DIM) comb[HDIM + t] = emb[cell[gi] * CE_DIM + t];
  __syncthreads();

  float y = bc[t];
  for (int k = 0; k < HDIM + CE_DIM; ++k) y += comb[k] * Wc[k * HDIM + t];
  float mu = blocksum128(y, red, t) * (1.0f / 128.0f);
  float dy = y - mu;
  float var = blocksum128(dy * dy, red, t) * (1.0f / 128.0f);
  y = dy * rsqrtf(var + LN_EPS) * lncw[t] + lncb[t];
  y = fmaxf(y, 0.0f);
  __syncthreads();
  vec[t] = y;
  __syncthreads();

  float z = bf1[t];
  for (int k = 0; k < HDIM; ++k) z += vec[k] * Wf1[k * HDIM + t];
  z = fmaxf(z, 0.0f);
  mu = blocksum128(z, red, t) * (1.0f / 128.0f);
  float dz = z - mu;
  var = blocksum128(dz * dz, red, t) * (1.0f / 128.0f);
  z = dz * rsqrtf(var + LN_EPS) * ln1w[t] + ln1b[t];
  __syncthreads();
  vec[t] = z;
  __syncthreads();

  float o = bf2[t];
  for (int k = 0; k < HDIM; ++k) o += vec[k] * Wf2[k * LDIM + t];
  out[gi * LDIM + t] = o;
}

extern "C" void kernel_launch(void* const* d_in, const int* in_sizes, int n_in,
                              void* d_out, int out_size, void* d_ws, size_t ws_size,
                              hipStream_t stream) {
  (void)in_sizes; (void)n_in; (void)out_size; (void)ws_size;

  const float* x    = (const float*)d_in[0];
  const int*   ei   = (const int*)d_in[1];
  const int*   batch= (const int*)d_in[2];
  const int*   cell = (const int*)d_in[3];
  const float* W1   = (const float*)d_in[4];
  const float* b1   = (const float*)d_in[5];
  const float* W2   = (const float*)d_in[6];
  const float* b2   = (const float*)d_in[7];
  const float* emb  = (const float*)d_in[8];
  const float* Wc   = (const float*)d_in[9];
  const float* bc   = (const float*)d_in[10];
  const float* lncw = (const float*)d_in[11];
  const float* lncb = (const float*)d_in[12];
  const float* Wf1  = (const float*)d_in[13];
  const float* bf1  = (const float*)d_in[14];
  const float* ln1w = (const float*)d_in[15];
  const float* ln1b = (const float*)d_in[16];
  const float* Wf2  = (const float*)d_in[17];
  const float* bf2  = (const float*)d_in[18];
  float* out = (float*)d_out;

  const int* src = ei;             // edge_index[0]
  const int* dst = ei + N_EDGES;   // edge_index[1]

  float* ws   = (float*)d_ws;
  float* deg  = ws;                                   // N floats (becomes dinv)
  float* bufA = ws + 50176;                           // N*H, 256B-aligned offset
  float* bufB = bufA + (size_t)N_NODES * HDIM;        // N*H
  float* gpool= bufB + (size_t)N_NODES * HDIM;        // G*H

  const int NH = N_NODES * HDIM;
  const int ntiles = N_NODES / 16;   // 3125, exact

  // symmetric-normalization degrees (self loop => init 1.0)
  set_f32<<<(N_NODES + 255) / 256, 256, 0, stream>>>(deg, 1.0f, N_NODES);
  edge_degree<<<(N_EDGES + 255) / 256, 256, 0, stream>>>(dst, deg, N_EDGES);
  rsqrt_inplace<<<(N_NODES + 255) / 256, 256, 0, stream>>>(deg, N_NODES);

  // layer 1: h = x @ W1 ; init with self-loop ; edge agg ; relu(+b1)
  gemm128_wmma<<<(ntiles + 7) / 8, 256, 0, stream>>>(x, W1, bufA, ntiles);
  selfloop_init<<<(NH + 255) / 256, 256, 0, stream>>>(bufA, deg, bufB, NH);
  edge_aggregate<<<(N_EDGES + 7) / 8, 256, 0, stream>>>(src, dst, deg, bufA, bufB, N_EDGES);
  bias_relu<<<(NH + 255) / 256, 256, 0, stream>>>(b1, bufB, NH);

  // layer 2
  gemm128_wmma<<<(ntiles + 7) / 8, 256, 0, stream>>>(bufB, W2, bufA, ntiles);
  selfloop_init<<<(NH + 255) / 256, 256, 0, stream>>>(bufA, deg, bufB, NH);
  edge_aggregate<<<(N_EDGES + 7) / 8, 256, 0, stream>>>(src, dst, deg, bufA, bufB, N_EDGES);
  bias_relu<<<(NH + 255) / 256, 256, 0, stream>>>(b2, bufB, NH);

  // global add pool
  set_f32<<<(G_GRAPHS * HDIM + 255) / 256, 256, 0, stream>>>(gpool, 0.0f, G_GRAPHS * HDIM);
  pool_add<<<(N_NODES + 7) / 8, 256, 0, stream>>>(bufB, batch, gpool, N_NODES);

  // fused MLP head
  head_kernel<<<G_GRAPHS, 128, 0, stream>>>(gpool, cell, emb, Wc, bc, lncw, lncb,
                                            Wf1, bf1, ln1w, ln1b, Wf2, bf2, out);
}